// YOLO_Loss_69140383531511
// MI455X (gfx1250) — compile-verified
//
#include <hip/hip_runtime.h>
#include <hip/hip_bf16.h>

// Problem constants (from the reference)
#define S 128
#define B 2
#define M 1024
#define C 80
#define P (S * S * B)        // 32768 boxes
#define NPROB (S * S * C)    // 1,310,720 prob elements (divisible by 4)

#define LAMBDA_COORD 5.0f
#define LAMBDA_NO_OBJ 0.5f

typedef __attribute__((ext_vector_type(2))) float v2f;
typedef __attribute__((ext_vector_type(8))) float v8f;

__global__ void yolo_zero_out(float* out) { out[0] = 0.0f; }

// nan_to_num(sqrt(x)) for all reachable inputs:
//   x >= 0 -> sqrt(x);  x < 0 or NaN -> 0   (fmaxf(NaN,0)=0, fmaxf(neg,0)=0)
__device__ __forceinline__ float safe_sqrt(float x) {
    return sqrtf(fmaxf(x, 0.0f));
}

__global__ __launch_bounds__(256) void yolo_loss_kernel(
    const float4* __restrict__ main_boxes,   // [M] of (cx,cy,w,h)
    const float4* __restrict__ pred_boxes,   // [P] of (cx,cy,w,h)
    const float*  __restrict__ conf,         // [P]
    const float4* __restrict__ gmp,          // grid_main_probs as float4, NPROB/4
    const float4* __restrict__ gpp,          // grid_pred_probs as float4, NPROB/4
    const int*    __restrict__ matched_main, // [P]
    const float*  __restrict__ matched_iou,  // [P]
    float* __restrict__ out)
{
    __shared__ float bsum;
    if (threadIdx.x == 0) bsum = 0.0f;
    __syncthreads();

    const int tid = (int)(blockIdx.x * blockDim.x + threadIdx.x);
    const int nthreads = (int)(gridDim.x * blockDim.x);

    float acc = 0.0f;

    // ---- prob loss: sum (a-b)^2 over NPROB elements, b128 loads ----
    const int n4 = NPROB / 4;
    for (int i = tid; i < n4; i += nthreads) {
        float4 a = gmp[i];
        float4 b = gpp[i];
        float d0 = a.x - b.x, d1 = a.y - b.y, d2 = a.z - b.z, d3 = a.w - b.w;
        acc += d0 * d0 + d1 * d1 + d2 * d2 + d3 * d3;
    }

    // ---- box loss over P boxes ----
    for (int p = tid; p < P; p += nthreads) {
        float4 pb = pred_boxes[p];       // b128
        float  cf = conf[p];
        int    mi = matched_main[p];
        float  l;
        if (mi >= 0) {
            float4 mb = main_boxes[mi];  // gathered GT box
            float dx = mb.x - pb.x;
            float dy = mb.y - pb.y;
            float xy = dx * dx + dy * dy;
            float dw = safe_sqrt(mb.z) - safe_sqrt(pb.z);
            float dh = safe_sqrt(mb.w) - safe_sqrt(pb.w);
            float wh = dw * dw + dh * dh;
            float dc = matched_iou[p] - cf;
            l = LAMBDA_COORD * (xy + wh) + dc * dc;
        } else {
            l = LAMBDA_NO_OBJ * cf * cf;
        }
        acc += l;
    }

    // ---- wave32 cross-lane reduction via V_WMMA_F32_16X16X4_F32 ----
    // A (16x4 f32, 2 VGPRs): lanes 0-15 supply A[m][0] (VGPR0) and A[m][1] (VGPR1);
    // lanes 16-31 supply A[m][2], A[m][3]. We put each lane's partial in its first
    // A register and 0 in the second, so rowsum_m = partial_m + partial_{m+16}.
    // B = ones -> D[m][n] = rowsum_m. D layout: lane 0's 8 D VGPRs = rowsums 0..7,
    // lane 16's 8 D VGPRs = rowsums 8..15  =>  lanes 0 and 16 together hold the
    // full 32-lane sum. EXEC is all-1s here (no divergence at this point).
    v2f a;
    a[0] = acc;
    a[1] = 0.0f;
    v2f bones;
    bones[0] = 1.0f;
    bones[1] = 1.0f;
    v8f cz = {};
    v8f d = __builtin_amdgcn_wmma_f32_16x16x4_f32(
        /*neg_a=*/false, a, /*neg_b=*/false, bones,
        /*c_mod=*/(short)0, cz, /*reuse_a=*/false, /*reuse_b=*/false);

    float s = d[0] + d[1] + d[2] + d[3] + d[4] + d[5] + d[6] + d[7];

    unsigned lane = threadIdx.x & 31u;
    if (lane == 0u || lane == 16u) {
        atomicAdd(&bsum, s);             // ds_add_f32
    }
    __syncthreads();
    if (threadIdx.x == 0) {
        atomicAdd(out, bsum);            // global_atomic_add_f32, one per block
    }
}

extern "C" void kernel_launch(void* const* d_in, const int* in_sizes, int n_in,
                              void* d_out, int out_size, void* d_ws, size_t ws_size,
                              hipStream_t stream) {
    (void)in_sizes; (void)n_in; (void)out_size; (void)d_ws; (void)ws_size;

    const float4* main_boxes   = (const float4*)d_in[0];
    const float4* pred_boxes   = (const float4*)d_in[1];
    const float*  conf         = (const float*)d_in[2];
    const float4* gmp          = (const float4*)d_in[3];
    const float4* gpp          = (const float4*)d_in[4];
    const int*    matched_main = (const int*)d_in[5];
    const float*  matched_iou  = (const float*)d_in[6];
    float*        out          = (float*)d_out;

    // Zero the accumulator each call (harness poisons d_out; atomics need 0 init).
    yolo_zero_out<<<1, 1, 0, stream>>>(out);

    // ~11.3 MB of reads, bandwidth-bound: 1024 blocks x 256 threads (8 waves/block,
    // wave32) grid-striding keeps plenty of b128 loads in flight for 23.3 TB/s HBM.
    yolo_loss_kernel<<<1024, 256, 0, stream>>>(
        main_boxes, pred_boxes, conf, gmp, gpp, matched_main, matched_iou, out);
}